// ShareLinear_5892695130656
// MI455X (gfx1250) — compile-verified
//
#include <hip/hip_runtime.h>

// out[b,f] = in[b,f] * weight[f] + bias[nd, f]
// Pure streaming op: 512 MB traffic, 134 MFLOP -> 0.26 FLOP/B, HBM-bound,
// ~22 us floor at 23.3 TB/s. Optimal form on MI455X:
//   - b128 non-temporal loads/stores for the stream-once input/output
//     (TH=NT keeps the 192MB L2 from being churned by 512MB of dead data)
//   - regular temporal loads for weight/bias (16KB each, reused 16384x),
//     cached in registers across 16 batch rows per block
//   - fully unrolled row loop: constant 16KB row stride folds into the
//     24-bit IOFFSET field -> zero per-iteration address arithmetic,
//     16 loads in flight per thread (LOADcnt budget is 63)
//   - no software prefetch: TLP (8 waves/block x 4096 blocks) covers
//     latency; prefetch would double load-path address traffic.

typedef float v4f __attribute__((ext_vector_type(4)));

namespace {
constexpr int F_DIM = 4096;
constexpr int BATCH = 16384;
constexpr int TPB = 256;                        // 8 waves (wave32)
constexpr int COLS_PER_BLOCK = TPB * 4;         // 1024 cols (float4/thread)
constexpr int CHUNKS = F_DIM / COLS_PER_BLOCK;  // 4
constexpr int ROWS_PER_BLOCK = 16;              // weight/bias reuse factor
}

__global__ __launch_bounds__(TPB) void ShareLinear_scale_bias_kernel(
    const float* __restrict__ in, const float* __restrict__ weight,
    const float* __restrict__ bias, const int* __restrict__ nd_ptr,
    float* __restrict__ out) {
  const int chunk = blockIdx.x;                  // 0..CHUNKS-1
  const int row0 = blockIdx.y * ROWS_PER_BLOCK;  // first batch row
  const int col = chunk * COLS_PER_BLOCK + (int)threadIdx.x * 4;

  // nb_distrib is device-resident (graph capture: no host readback).
  // Uniform address -> scalar load.
  const int nd = *nd_ptr;
  const float* __restrict__ brow = bias + (size_t)nd * F_DIM;

  // Parameters: tiny, hot, reused by every batch row -> regular temporal
  // loads, then held in registers for all 16 rows.
  const v4f w4 = *(const v4f*)(weight + col);
  const v4f b4 = *(const v4f*)(brow + col);

  const float* __restrict__ ip = in + (size_t)row0 * F_DIM + col;
  float* __restrict__ op = out + (size_t)row0 * F_DIM + col;

#pragma unroll
  for (int r = 0; r < ROWS_PER_BLOCK; ++r) {
    // r*F_DIM*4B <= 245760 -> folds into the 24-bit instruction offset.
    v4f x = __builtin_nontemporal_load((const v4f*)(ip + r * F_DIM));
    v4f y = x * w4 + b4;  // 4x f32 fma
    __builtin_nontemporal_store(y, (v4f*)(op + r * F_DIM));
  }
}

extern "C" void kernel_launch(void* const* d_in, const int* in_sizes, int n_in,
                              void* d_out, int out_size, void* d_ws,
                              size_t ws_size, hipStream_t stream) {
  (void)in_sizes; (void)n_in; (void)out_size; (void)d_ws; (void)ws_size;
  const float* in = (const float*)d_in[0];
  const float* weight = (const float*)d_in[1];
  const float* bias = (const float*)d_in[2];
  const int* nd = (const int*)d_in[3];
  float* out = (float*)d_out;

  dim3 grid(CHUNKS, BATCH / ROWS_PER_BLOCK, 1);  // (4, 1024)
  dim3 block(TPB, 1, 1);
  ShareLinear_scale_bias_kernel<<<grid, block, 0, stream>>>(in, weight, bias,
                                                            nd, out);
}